// SurvivalGNN_16647293239617
// MI455X (gfx1250) — compile-verified
//
#include <hip/hip_runtime.h>

typedef float v2f __attribute__((ext_vector_type(2)));
typedef float v8f __attribute__((ext_vector_type(8)));

#define HDIM 64

// ---------- degree / normalization ----------
__global__ void k_init_deg(float* deg, int N) {
    int i = blockIdx.x * blockDim.x + threadIdx.x;
    if (i < N) deg[i] = 1.0f;  // self-loop contributes 1 to every node's degree
}

__global__ void k_deg_accum(const long long* __restrict__ dst, float* deg, int E) {
    int e = blockIdx.x * blockDim.x + threadIdx.x;
    if (e < E) atomicAdd(&deg[(int)dst[e]], 1.0f);
}

__global__ void k_dinv_inplace(float* deg, int N) {
    int i = blockIdx.x * blockDim.x + threadIdx.x;
    if (i < N) deg[i] = rsqrtf(deg[i]);  // deg >= 1 always (self-loops)
}

__global__ void k_zero(float* p, int n) {
    int i = blockIdx.x * blockDim.x + threadIdx.x;
    if (i < n) p[i] = 0.0f;
}

// ---------- WMMA GEMM: Out[M x 64] = X[M x K] @ W[K x 64] ----------
// One wave per 16-row tile; 4 column tiles (64 cols) held in 4 v8f accumulators.
// V_WMMA_F32_16X16X4_F32, fp32-exact. M must be a multiple of 16 (100000 = 6250*16).
__global__ void k_gemm_wmma(const float* __restrict__ X, const float* __restrict__ W,
                            float* __restrict__ Out, int K, int rowTiles) {
    int wave = blockIdx.x * (blockDim.x >> 5) + (threadIdx.x >> 5);
    if (wave >= rowTiles) return;           // wave-uniform: EXEC stays all-ones for WMMA
    int lane  = threadIdx.x & 31;
    int row0  = wave * 16;
    int m     = lane & 15;
    int khalf = (lane >> 4) * 2;            // lanes 0-15 -> K0/K1, lanes 16-31 -> K2/K3

    v8f c0 = {}, c1 = {}, c2 = {}, c3 = {};
    const float* xrow = X + (size_t)(row0 + m) * K;
    int nc = lane & 15;

    for (int k = 0; k < K; k += 4) {
        v2f a;
        a.x = xrow[k + khalf];
        a.y = xrow[k + khalf + 1];
        const float* w0 = W + (size_t)(k + khalf) * HDIM;
        const float* w1 = w0 + HDIM;
        v2f b0, b1v, b2v, b3v;
        b0.x  = w0[nc];      b0.y  = w1[nc];
        b1v.x = w0[nc + 16]; b1v.y = w1[nc + 16];
        b2v.x = w0[nc + 32]; b2v.y = w1[nc + 32];
        b3v.x = w0[nc + 48]; b3v.y = w1[nc + 48];
        c0 = __builtin_amdgcn_wmma_f32_16x16x4_f32(false, a, false, b0,  (short)0, c0, false, false);
        c1 = __builtin_amdgcn_wmma_f32_16x16x4_f32(false, a, false, b1v, (short)0, c1, false, false);
        c2 = __builtin_amdgcn_wmma_f32_16x16x4_f32(false, a, false, b2v, (short)0, c2, false, false);
        c3 = __builtin_amdgcn_wmma_f32_16x16x4_f32(false, a, false, b3v, (short)0, c3, false, false);
    }

    // C/D layout: VGPR i, lanes 0-15 -> M=i ; lanes 16-31 -> M=8+i ; N = lane&15
    int rbase = row0 + ((lane >> 4) << 3);
    int col   = lane & 15;
    for (int i = 0; i < 8; ++i) {
        float* o = Out + (size_t)(rbase + i) * HDIM + col;
        o[0]  = c0[i];
        o[16] = c1[i];
        o[32] = c2[i];
        o[48] = c3[i];
    }
}

// ---------- edge aggregation: agg[dst] += h[src] * dinv[src]*dinv[dst] ----------
// One wave per edge; each lane handles 2 consecutive features (64 = 32*2).
__global__ void k_edge_agg(const long long* __restrict__ src, const long long* __restrict__ dst,
                           const float* __restrict__ hp, const float* __restrict__ dinv,
                           float* __restrict__ agg, int E) {
    unsigned t = blockIdx.x * blockDim.x + threadIdx.x;
    int e = (int)(t >> 5);
    if (e >= E) return;
    int lane = (int)(t & 31u);
    int s = (int)src[e];
    int d = (int)dst[e];
    float nrm = dinv[s] * dinv[d];
    const float2 v = *((const float2*)(hp + (size_t)s * HDIM) + lane);
    float* o = agg + (size_t)d * HDIM + lane * 2;
    atomicAdd(o,     v.x * nrm);
    atomicAdd(o + 1, v.y * nrm);
}

// ---------- fused self-loop + bias + ReLU (in place on agg) ----------
__global__ void k_relu_bias_self(float* __restrict__ agg, const float* __restrict__ hp,
                                 const float* __restrict__ dinv, const float* __restrict__ bias,
                                 int total) {
    int i = blockIdx.x * blockDim.x + threadIdx.x;
    if (i >= total) return;
    int n = i >> 6;
    int c = i & 63;
    float di = dinv[n];
    float v = agg[i] + hp[i] * di * di + bias[c];
    agg[i] = v > 0.0f ? v : 0.0f;
}

// ---------- heads: survival/event = h2 @ W + b, one wave per node ----------
__global__ void k_heads(const float* __restrict__ h,
                        const float* __restrict__ Wt, const float* __restrict__ bt,
                        const float* __restrict__ We, const float* __restrict__ be,
                        float* __restrict__ out, int N) {
    int node = blockIdx.x * (blockDim.x >> 5) + (threadIdx.x >> 5);
    if (node >= N) return;
    int lane = threadIdx.x & 31;
    const float* hr = h + (size_t)node * HDIM;
    float h0 = hr[lane], h1 = hr[lane + 32];
    float st = h0 * Wt[lane] + h1 * Wt[lane + 32];
    float ev = h0 * We[lane] + h1 * We[lane + 32];
    for (int off = 16; off > 0; off >>= 1) {
        st += __shfl_down(st, off, 32);
        ev += __shfl_down(ev, off, 32);
    }
    if (lane == 0) {
        out[node]     = st + bt[0];
        out[N + node] = ev + be[0];
    }
}

extern "C" void kernel_launch(void* const* d_in, const int* in_sizes, int n_in,
                              void* d_out, int out_size, void* d_ws, size_t ws_size,
                              hipStream_t stream) {
    const float*     x    = (const float*)d_in[0];
    const long long* eidx = (const long long*)d_in[1];
    const float*     W1   = (const float*)d_in[2];
    const float*     b1   = (const float*)d_in[3];
    const float*     W2   = (const float*)d_in[4];
    const float*     b2   = (const float*)d_in[5];
    const float*     Wt   = (const float*)d_in[6];
    const float*     bt   = (const float*)d_in[7];
    const float*     We   = (const float*)d_in[8];
    const float*     be   = (const float*)d_in[9];

    int N = in_sizes[0] / 128;   // 100000
    int E = in_sizes[1] / 2;     // 3200000
    const long long* srcs = eidx;
    const long long* dsts = eidx + E;

    // workspace layout: dinv[N] | A[N*64] | B[N*64]   (~52 MB)
    float* dinv = (float*)d_ws;
    float* A    = dinv + N;              // h_pre buffer
    float* B    = A + (size_t)N * HDIM;  // agg / h buffer
    float* out  = (float*)d_out;

    const int T = 256;
    int totNH       = N * HDIM;
    int rowTiles    = N / 16;                 // 6250
    int gemmBlocks  = (rowTiles + 7) / 8;     // 8 waves per 256-thread block
    unsigned eThreads = (unsigned)E * 32u;    // one wave per edge

    // normalization: deg -> dinv (in place)
    k_init_deg<<<(N + T - 1) / T, T, 0, stream>>>(dinv, N);
    k_deg_accum<<<(E + T - 1) / T, T, 0, stream>>>(dsts, dinv, E);
    k_dinv_inplace<<<(N + T - 1) / T, T, 0, stream>>>(dinv, N);

    // layer 1: A = x @ W1 ; B = scatter(A) ; B = relu(B + self + b1)
    k_gemm_wmma<<<gemmBlocks, T, 0, stream>>>(x, W1, A, 128, rowTiles);
    k_zero<<<(totNH + T - 1) / T, T, 0, stream>>>(B, totNH);
    k_edge_agg<<<(eThreads + T - 1) / T, T, 0, stream>>>(srcs, dsts, A, dinv, B, E);
    k_relu_bias_self<<<(totNH + T - 1) / T, T, 0, stream>>>(B, A, dinv, b1, totNH);

    // layer 2: A = B @ W2 ; B = scatter(A) ; B = relu(B + self + b2)
    k_gemm_wmma<<<gemmBlocks, T, 0, stream>>>(B, W2, A, HDIM, rowTiles);
    k_zero<<<(totNH + T - 1) / T, T, 0, stream>>>(B, totNH);
    k_edge_agg<<<(eThreads + T - 1) / T, T, 0, stream>>>(srcs, dsts, A, dinv, B, E);
    k_relu_bias_self<<<(totNH + T - 1) / T, T, 0, stream>>>(B, A, dinv, b2, totNH);

    // heads
    k_heads<<<(N + 7) / 8, T, 0, stream>>>(B, Wt, bt, We, be, out, N);
}